// ExponentialTimeDiffEmbedding_30262339567984
// MI455X (gfx1250) — compile-verified
//
#include <hip/hip_runtime.h>
#include <stdint.h>

// Problem constants (from the reference)
#define CLIP   256
#define HIDDEN 64
#define BATCH  32
#define SEQ    256

// Tiling: each gather block owns ITILE consecutive 'i' rows of one batch.
#define ITILE 4
#define EMB_CHUNKS     (((CLIP + 1) * HIDDEN) / 4)        // 4112 float4 chunks
#define SMEM_EMB_BYTES ((CLIP + 1) * HIDDEN * 4)          // 65792 B
#define SMEM_BYTES     (SMEM_EMB_BYTES + ITILE * SEQ * 4) // + 4KB idx table

typedef __attribute__((ext_vector_type(4))) float v4f;

// ---------------------------------------------------------------------------
// Kernel 1: per-batch minimum nonzero pairwise |t_i - t_j|  (sentinel INT_MAX)
// One 256-thread block per batch; trivial cost vs. the 512MB gather.
// ---------------------------------------------------------------------------
__global__ __launch_bounds__(256) void tisas_tmin_kernel(
    const int* __restrict__ ts, int* __restrict__ tmin) {
  __shared__ int      sts[SEQ];
  __shared__ unsigned sred[SEQ];
  const int b   = blockIdx.x;
  const int tid = threadIdx.x;
  sts[tid] = ts[b * SEQ + tid];
  __syncthreads();
  const int ti = sts[tid];
  unsigned m = 0x7fffffffu;  // int32 max sentinel, matches reference
#pragma unroll 8
  for (int j = 0; j < SEQ; ++j) {
    int d = ti - sts[j];
    unsigned ad = (unsigned)(d < 0 ? -d : d);
    if (ad != 0u && ad < m) m = ad;
  }
  sred[tid] = m;
  __syncthreads();
  for (int s = 128; s > 0; s >>= 1) {
    if (tid < s) sred[tid] = min(sred[tid], sred[tid + s]);
    __syncthreads();
  }
  if (tid == 0) tmin[b] = (int)sred[0];
}

// ---------------------------------------------------------------------------
// Kernel 2: stage the 257x64 f32 table into LDS via the CDNA5 async
// global->LDS path, build the idx table, then stream 256B embedding rows
// straight from LDS to global memory with async NT stores — data never
// touches VGPRs; the wave only does address math while the async engine
// saturates the 23.3 TB/s store path.
// Grid: BATCH * (SEQ/ITILE) blocks of 256 threads (8 wave32).
// ---------------------------------------------------------------------------
__global__ __launch_bounds__(256) void tisas_gather_kernel(
    const int* __restrict__ ts, const float* __restrict__ emb,
    const int* __restrict__ tmin, float* __restrict__ out) {
  extern __shared__ char smem_raw[];
  v4f* semb = (v4f*)smem_raw;                       // 4112 x float4 (257 rows)
  int* sidx = (int*)(smem_raw + SMEM_EMB_BYTES);    // ITILE*SEQ indices

  const int tid = threadIdx.x;
  const int blocks_per_batch = SEQ / ITILE;         // 64
  const int b  = blockIdx.x / blocks_per_batch;
  const int i0 = (blockIdx.x % blocks_per_batch) * ITILE;

  // --- Async-stage the whole embedding table into LDS (ASYNCcnt path). ---
  // 4112 chunks / 256 threads; clamp tail so every wave issues with full EXEC.
  for (int k = tid; k < ((EMB_CHUNKS + 255) & ~255); k += 256) {
    const int kc = (k < EMB_CHUNKS) ? k : (EMB_CHUNKS - 1);
    const unsigned lds_addr = (unsigned)(size_t)(semb + kc);  // LDS byte addr
    const unsigned goff     = (unsigned)kc * 16u;             // global byte off
    asm volatile("global_load_async_to_lds_b128 %0, %1, %2"
                 :: "v"(lds_addr), "v"(goff), "s"(emb)
                 : "memory");
  }

  // --- Overlap: compute idx[ii][j] = min(|t_{i0+ii} - t_j| / t_min, CLIP). ---
  const unsigned tm = (unsigned)tmin[b];
  for (int k = tid; k < ITILE * SEQ; k += 256) {
    const int ii = k >> 8;
    const int j  = k & (SEQ - 1);
    const int d  = ts[b * SEQ + (i0 + ii)] - ts[b * SEQ + j];
    const unsigned ad = (unsigned)(d < 0 ? -d : d);
    const unsigned q  = ad / tm;  // tm >= 1 (sentinel case => ad == 0)
    sidx[k] = (q > (unsigned)CLIP) ? CLIP : (int)q;
  }

  // All waves staged their own chunks: wait own ASYNCcnt, then barrier so the
  // whole table + idx table are visible to everyone.
  asm volatile("s_wait_asynccnt 0" ::: "memory");
  __syncthreads();

  // --- Stream rows LDS -> global via async NT b128 stores (no VGPR data). ---
  // 16 lanes cover one 256B row; a wave covers two consecutive j rows, so its
  // 32 lanes emit 512B of contiguous global addresses per iteration.
  const int lane16 = tid & 15;
  const int rowgrp = tid >> 4;  // 0..15: which j-row (mod 16) this thread serves
#pragma unroll
  for (int ii = 0; ii < ITILE; ++ii) {
    // Byte offset of row block [b, i0+ii, 0, 0] in out (max 2^29 < 2^31).
    const unsigned obase =
        ((unsigned)(b * SEQ + (i0 + ii)) * SEQ) * (HIDDEN * 4u);
#pragma unroll 4
    for (int it = 0; it < 16; ++it) {
      const int j   = it * 16 + rowgrp;
      const int idx = sidx[ii * SEQ + j];
      const unsigned lds_src =
          (unsigned)(size_t)(semb + idx * 16 + lane16);         // LDS byte addr
      const unsigned goff = obase + (unsigned)j * (HIDDEN * 4u)
                                  + (unsigned)lane16 * 16u;     // global offset
      asm volatile("global_store_async_from_lds_b128 %0, %1, %2 th:TH_STORE_NT"
                   :: "v"(goff), "v"(lds_src), "s"(out)
                   : "memory");
    }
  }

  // Drain outstanding async stores before the wave retires (S_ENDPGM's
  // implicit wait-idle also covers this; explicit for clarity).
  asm volatile("s_wait_asynccnt 0" ::: "memory");
}

// ---------------------------------------------------------------------------
extern "C" void kernel_launch(void* const* d_in, const int* in_sizes, int n_in,
                              void* d_out, int out_size, void* d_ws, size_t ws_size,
                              hipStream_t stream) {
  (void)in_sizes; (void)n_in; (void)out_size; (void)ws_size;
  const int*   ts  = (const int*)d_in[0];     // [32,256] int32
  const float* emb = (const float*)d_in[1];   // [257,64] float32
  float*       out = (float*)d_out;           // [32,256,256,64] float32
  int*         tmn = (int*)d_ws;              // 32 ints of scratch

  // Allow ~70KB of dynamic LDS per workgroup (WGP has 320KB).
  (void)hipFuncSetAttribute((const void*)tisas_gather_kernel,
                            hipFuncAttributeMaxDynamicSharedMemorySize,
                            SMEM_BYTES);

  tisas_tmin_kernel<<<BATCH, 256, 0, stream>>>(ts, tmn);
  tisas_gather_kernel<<<BATCH * (SEQ / ITILE), 256, SMEM_BYTES, stream>>>(
      ts, emb, tmn, out);
}